// FractionalUpsample_58076547776586
// MI455X (gfx1250) — compile-verified
//
#include <hip/hip_runtime.h>
#include <hip/hip_bf16.h>

typedef __attribute__((ext_vector_type(2))) float v2f;
typedef __attribute__((ext_vector_type(8))) float v8f;

// Problem dims (fixed by the reference)
#define BB 16
#define HH 128
#define WW 128
#define CC 64
#define FF 64
#define OH 129
#define OW 129
#define RPITCH 66   // 64 + 2 floats padding -> avoids 16-way LDS bank conflicts

__global__ __launch_bounds__(256)
void frac_resize_conv_wmma(const float* __restrict__ x,
                           const float* __restrict__ kmat,
                           const float* __restrict__ bias,
                           float* __restrict__ out)
{
    __shared__ float sK[CC * FF];          // conv kernel, [c][f] row-major, 16 KB
    __shared__ float sR[OW * RPITCH];      // resampled row, [j][c] pitch 66, ~34 KB

    const int b   = blockIdx.y;
    const int i   = blockIdx.x;
    const int tid = threadIdx.x;

    // ---- stage conv kernel into LDS (coalesced) ----
    for (int t = tid; t < CC * FF; t += 256) sK[t] = kmat[t];

    // ---- analytic 2-tap vertical weights for output row i ----
    const int   h0  = (i == 0) ? 0 : (i - 1);
    const int   h1  = (h0 + 1 < HH) ? (h0 + 1) : (HH - 1);
    const float wh0 = (i == 0) ? 1.0f : (float)i * (1.0f / 128.0f);
    const float wh1 = 1.0f - wh0;

    const float* __restrict__ xr0 = x + (size_t)(b * HH + h0) * (WW * CC);
    const float* __restrict__ xr1 = x + (size_t)(b * HH + h1) * (WW * CC);

    // ---- build fully-resampled row R[j][c] in LDS (x rows are L2-resident) ----
    for (int t = tid; t < OW * CC; t += 256) {
        const int   j   = t >> 6;
        const int   c   = t & 63;
        const int   w0  = (j == 0) ? 0 : (j - 1);
        const int   w1  = (w0 + 1 < WW) ? (w0 + 1) : (WW - 1);
        const float ww0 = (j == 0) ? 1.0f : (float)j * (1.0f / 128.0f);
        const float ww1 = 1.0f - ww0;
        const float v0  = wh0 * xr0[w0 * CC + c] + wh1 * xr1[w0 * CC + c];
        const float v1  = wh0 * xr0[w1 * CC + c] + wh1 * xr1[w1 * CC + c];
        sR[j * RPITCH + c] = ww0 * v0 + ww1 * v1;
    }
    __syncthreads();

    // ---- WMMA GEMM: (129 x 64) = R(129 x 64) @ K(64 x 64) + bias ----
    const int wave = tid >> 5;   // 0..7
    const int lane = tid & 31;
    const int half = lane >> 4;  // 0: K lo pair / M rows 0-7 ; 1: K hi pair / M rows 8-15
    const int l16  = lane & 15;

    for (int mt = wave; mt < 9; mt += 8) {          // 9 tiles of 16 j-positions
        const int j0 = mt * 16;
        int jA = j0 + l16;                           // A-fragment row (clamped; masked on store)
        if (jA > OW - 1) jA = OW - 1;

        v8f acc[4];
        #pragma unroll
        for (int nt = 0; nt < 4; ++nt) {
            const float bv = bias[nt * 16 + l16];    // bias depends only on f (N column)
            #pragma unroll
            for (int v = 0; v < 8; ++v) acc[nt][v] = bv;
        }

        #pragma unroll
        for (int kk = 0; kk < 16; ++kk) {            // K = 64 channels, 4 per wmma
            const int c0 = kk * 4 + half * 2;
            v2f a;
            a.x = sR[jA * RPITCH + c0];              // A: M=l16, K={c0, c0+1}
            a.y = sR[jA * RPITCH + c0 + 1];
            #pragma unroll
            for (int nt = 0; nt < 4; ++nt) {
                const int f = nt * 16 + l16;
                v2f bm;
                bm.x = sK[c0 * FF + f];              // B: K=c0 row, N=l16
                bm.y = sK[(c0 + 1) * FF + f];
                acc[nt] = __builtin_amdgcn_wmma_f32_16x16x4_f32(
                    /*neg_a=*/false, a, /*neg_b=*/false, bm,
                    /*c_mod=*/(short)0, acc[nt],
                    /*reuse_a=*/false, /*reuse_b=*/false);
            }
        }

        // ---- store D: VGPR v -> row (v + 8*half), lane l16 -> column ----
        #pragma unroll
        for (int v = 0; v < 8; ++v) {
            const int j = j0 + v + half * 8;
            if (j < OW) {
                const size_t base = (((size_t)(b * OH + i)) * OW + j) * FF;
                #pragma unroll
                for (int nt = 0; nt < 4; ++nt) {
                    out[base + nt * 16 + l16] = acc[nt][v];
                }
            }
        }
    }
}

extern "C" void kernel_launch(void* const* d_in, const int* in_sizes, int n_in,
                              void* d_out, int out_size, void* d_ws, size_t ws_size,
                              hipStream_t stream) {
    const float* x    = (const float*)d_in[0];  // (16,128,128,64)
    const float* kmat = (const float*)d_in[1];  // (64,64)
    const float* bias = (const float*)d_in[2];  // (64,)
    float* out = (float*)d_out;                 // (16,129,129,64)

    dim3 grid(OH, BB);   // one workgroup per (output row i, batch b)
    dim3 block(256);     // 8 wave32 waves
    frac_resize_conv_wmma<<<grid, block, 0, stream>>>(x, kmat, bias, out);
}